// M2M_18734647345158
// MI455X (gfx1250) — compile-verified
//
#include <hip/hip_runtime.h>
#include <hip/hip_bf16.h>

#define N_MAP      128
#define N_NODES    100000
#define NUM_SCALES 6
#define E_SCALE    100000
#define E_LR       20000
#define EPSV       1e-5f
#define LDSTRIDE   132   // 132 mod 64 banks = 4 -> conflict-free b64 lane pattern
#define T_PER_BLK  10    // 16-edge/row tiles per block (6250/10, 1250/10 both exact)

typedef __attribute__((ext_vector_type(2))) float v2f;
typedef __attribute__((ext_vector_type(8))) float v8f;

#if defined(__has_builtin)
#  if __has_builtin(__builtin_amdgcn_sched_barrier)
#    define SCHED_FENCE() __builtin_amdgcn_sched_barrier(0)
#  else
#    define SCHED_FENCE()
#  endif
#else
#  define SCHED_FENCE()
#endif

// ---------------------------------------------------------------------------
// Per-wave B-fragment preload: wave owns output columns [n0, n0+16).
// B[k][n] = W[n0+n][k]; lane(hi, m): wb[t] = { W[(n0+m)*128 + 4t+2hi], ...+1 }
// 64 VGPRs, loaded once per block, reused for all tiles.
// ---------------------------------------------------------------------------
__device__ __forceinline__ void preload_w(v2f (&wb)[32], const float* __restrict__ W,
                                          int n0, int m, int hi) {
#pragma unroll
    for (int k = 0; k < 32; ++k) {
        const float* wp = W + (n0 + m) * N_MAP + 4 * k + 2 * hi;
        wb[k].x = wp[0];
        wb[k].y = wp[1];
    }
}

// 16x16x128 tile against register-resident B.
// A fragments pipelined in 4 groups of 8 (ping-pong register groups): group
// g+1's 4 ds_load_2addr_b64 issue before group g's 8 WMMAs (sched_barrier
// enforced), so WMMAs wait with nonzero dscnt headroom. Only 32 A VGPRs in
// flight -> no spills (total ~124 VGPRs with W + accum).
__device__ __forceinline__ v8f tile_mma_f32(const float* __restrict__ As,
                                            const v2f (&wb)[32], int m, int hi) {
    const float* ap = As + m * LDSTRIDE + 2 * hi;
    v2f aA[8], aB[8];
#pragma unroll
    for (int j = 0; j < 8; ++j) {
        aA[j].x = ap[4 * j];
        aA[j].y = ap[4 * j + 1];
    }
    v8f c = {};
#pragma unroll
    for (int g = 0; g < 4; ++g) {
        if (g < 3) {
            v2f (&nxt)[8] = (g & 1) ? aA : aB;
#pragma unroll
            for (int j = 0; j < 8; ++j) {
                nxt[j].x = ap[4 * (8 * (g + 1) + j)];
                nxt[j].y = ap[4 * (8 * (g + 1) + j) + 1];
            }
        }
        SCHED_FENCE();   // next group's DS loads are in flight before this group's MMAs
        v2f (&curg)[8] = (g & 1) ? aB : aA;
#pragma unroll
        for (int j = 0; j < 8; ++j)
            c = __builtin_amdgcn_wmma_f32_16x16x4_f32(false, curg[j], false, wb[8 * g + j],
                                                      (short)0, c, false, false);
    }
    return c;
}

// ---------------------------------------------------------------------------
// Dense GEMM: out = A @ W^T. Block = T_PER_BLK row tiles of 16, double-buffered
// LDS staging, W fragments in registers. 8 waves = 8 column tiles.
// ---------------------------------------------------------------------------
__global__ void gemm_nt_kernel(float* __restrict__ out,
                               const float* __restrict__ A,
                               const float* __restrict__ W) {
    __shared__ float As[2][16 * LDSTRIDE];

    const int t = threadIdx.x;
    const int wave = t >> 5, lane = t & 31;
    const int m = lane & 15, hi = lane >> 4;
    const int n0 = wave * 16;
    const int r = t >> 4, cseg = (t & 15) * 8;

    v2f wb[32];
    preload_w(wb, W, n0, m, hi);

    int row0 = blockIdx.x * (T_PER_BLK * 16);
    // prologue: stage tile 0
    float4 p0, p1;
    {
        const float* srow = A + (size_t)(row0 + r) * N_MAP + cseg;
        p0 = *(const float4*)(srow);
        p1 = *(const float4*)(srow + 4);
    }
    *(float4*)&As[0][r * LDSTRIDE + cseg]     = p0;
    *(float4*)&As[0][r * LDSTRIDE + cseg + 4] = p1;

    for (int it = 0; it < T_PER_BLK; ++it) {
        __syncthreads();
        const int cur = it & 1;
        // prefetch next tile into registers while MMA runs
        if (it + 1 < T_PER_BLK) {
            const float* srow = A + (size_t)(row0 + 16 + r) * N_MAP + cseg;
            p0 = *(const float4*)(srow);
            p1 = *(const float4*)(srow + 4);
        }
        v8f c = tile_mma_f32(&As[cur][0], wb, m, hi);
#pragma unroll
        for (int rr = 0; rr < 8; ++rr)
            out[(size_t)(row0 + 8 * hi + rr) * N_MAP + n0 + m] = c[rr];
        if (it + 1 < T_PER_BLK) {
            const int nxt = cur ^ 1;
            *(float4*)&As[nxt][r * LDSTRIDE + cseg]     = p0;
            *(float4*)&As[nxt][r * LDSTRIDE + cseg + 4] = p1;
            row0 += 16;
        }
    }
}

// ---------------------------------------------------------------------------
// All 14 edge sets in one grid (blocks-per-set: 625 scale / 125 LR):
//   blocks [0, 7500): scale sets, set = b/625, s = set>>1, even=pre odd=suc
//   blocks [7500, 7750): left (125) then right (125)
// Per tile: gather 16 source rows -> LDS (double-buffered), wave WMMA against
// register-resident W, atomic f32 scatter-add to destination rows.
// ---------------------------------------------------------------------------
__global__ void edge_kernel(float* __restrict__ temp,
                            const float* __restrict__ feat,
                            const float* __restrict__ Wp,  // [6][128][128] (this iter)
                            const float* __restrict__ Ws,  // [6][128][128]
                            const float* __restrict__ Wl,
                            const float* __restrict__ Wr,
                            const int* __restrict__ pre_u, const int* __restrict__ pre_v,
                            const int* __restrict__ suc_u, const int* __restrict__ suc_v,
                            const int* __restrict__ left_u, const int* __restrict__ left_v,
                            const int* __restrict__ right_u, const int* __restrict__ right_v) {
    __shared__ float As[2][16 * LDSTRIDE];
    __shared__ int   us[2][16];

    const int BS = (E_SCALE / 16) / T_PER_BLK;  // 625
    const int BL = (E_LR / 16) / T_PER_BLK;     // 125
    const int b = blockIdx.x;

    const int* uPtr; const int* vPtr; const float* W; int e0;
    if (b < 12 * BS) {
        const int set = b / BS;
        const int blk = b - set * BS;
        const int s = set >> 1;
        if ((set & 1) == 0) { uPtr = pre_u + s * E_SCALE; vPtr = pre_v + s * E_SCALE; W = Wp + s * N_MAP * N_MAP; }
        else                { uPtr = suc_u + s * E_SCALE; vPtr = suc_v + s * E_SCALE; W = Ws + s * N_MAP * N_MAP; }
        e0 = blk * (T_PER_BLK * 16);
    } else {
        const int idx = b - 12 * BS;
        const int set = idx / BL;
        const int blk = idx - set * BL;
        if (set == 0) { uPtr = left_u;  vPtr = left_v;  W = Wl; }
        else          { uPtr = right_u; vPtr = right_v; W = Wr; }
        e0 = blk * (T_PER_BLK * 16);
    }

    const int t = threadIdx.x;
    const int wave = t >> 5, lane = t & 31;
    const int m = lane & 15, hi = lane >> 4;
    const int n0 = wave * 16;
    const int r = t >> 4, cseg = (t & 15) * 8;

    v2f wb[32];
    preload_w(wb, W, n0, m, hi);

    // prologue: gather tile 0
    float4 p0, p1; int uval = 0;
    {
        const int src = vPtr[e0 + r];
        const float* srow = feat + (size_t)src * N_MAP + cseg;
        p0 = *(const float4*)(srow);
        p1 = *(const float4*)(srow + 4);
        if (t < 16) uval = uPtr[e0 + t];
    }
    *(float4*)&As[0][r * LDSTRIDE + cseg]     = p0;
    *(float4*)&As[0][r * LDSTRIDE + cseg + 4] = p1;
    if (t < 16) us[0][t] = uval;

    for (int it = 0; it < T_PER_BLK; ++it) {
        __syncthreads();
        const int cur = it & 1;
        // prefetch next tile's gather into registers while MMA runs
        if (it + 1 < T_PER_BLK) {
            const int src = vPtr[e0 + 16 + r];
            const float* srow = feat + (size_t)src * N_MAP + cseg;
            p0 = *(const float4*)(srow);
            p1 = *(const float4*)(srow + 4);
            if (t < 16) uval = uPtr[e0 + 16 + t];
        }
        v8f c = tile_mma_f32(&As[cur][0], wb, m, hi);
#pragma unroll
        for (int rr = 0; rr < 8; ++rr) {
            const int dst = us[cur][8 * hi + rr];
            unsafeAtomicAdd(&temp[(size_t)dst * N_MAP + n0 + m], c[rr]);
        }
        if (it + 1 < T_PER_BLK) {
            const int nxt = cur ^ 1;
            *(float4*)&As[nxt][r * LDSTRIDE + cseg]     = p0;
            *(float4*)&As[nxt][r * LDSTRIDE + cseg + 4] = p1;
            if (t < 16) us[nxt][t] = uval;
            e0 += 16;
        }
    }
}

// ---------------------------------------------------------------------------
// GroupNorm(1,C) over channel dim per row + ReLU. One wave32 per row (4 ch/lane).
// ---------------------------------------------------------------------------
__global__ void gn_relu_kernel(float* __restrict__ out, const float* __restrict__ in,
                               const float* __restrict__ w, const float* __restrict__ b) {
    const int t = threadIdx.x;
    const int wave = t >> 5, lane = t & 31;
    const long row = (long)blockIdx.x * 8 + wave;
    if (row >= N_NODES) return;
    float4 v = *(const float4*)(in + row * N_MAP + lane * 4);
    float s = v.x + v.y + v.z + v.w;
#pragma unroll
    for (int off = 16; off; off >>= 1) s += __shfl_xor(s, off);
    const float mean = s * (1.0f / N_MAP);
    const float4 d = {v.x - mean, v.y - mean, v.z - mean, v.w - mean};
    float q = d.x * d.x + d.y * d.y + d.z * d.z + d.w * d.w;
#pragma unroll
    for (int off = 16; off; off >>= 1) q += __shfl_xor(q, off);
    const float inv = rsqrtf(q * (1.0f / N_MAP) + EPSV);
    const float4 w4 = *(const float4*)(w + lane * 4);
    const float4 b4 = *(const float4*)(b + lane * 4);
    float4 o;
    o.x = fmaxf(d.x * inv * w4.x + b4.x, 0.0f);
    o.y = fmaxf(d.y * inv * w4.y + b4.y, 0.0f);
    o.z = fmaxf(d.z * inv * w4.z + b4.z, 0.0f);
    o.w = fmaxf(d.w * inv * w4.w + b4.w, 0.0f);
    *(float4*)(out + row * N_MAP + lane * 4) = o;
}

// GroupNorm (no act) + residual add + ReLU.
__global__ void gn_res_relu_kernel(float* __restrict__ out, const float* __restrict__ in,
                                   const float* __restrict__ res,
                                   const float* __restrict__ w, const float* __restrict__ b) {
    const int t = threadIdx.x;
    const int wave = t >> 5, lane = t & 31;
    const long row = (long)blockIdx.x * 8 + wave;
    if (row >= N_NODES) return;
    float4 v = *(const float4*)(in + row * N_MAP + lane * 4);
    float s = v.x + v.y + v.z + v.w;
#pragma unroll
    for (int off = 16; off; off >>= 1) s += __shfl_xor(s, off);
    const float mean = s * (1.0f / N_MAP);
    const float4 d = {v.x - mean, v.y - mean, v.z - mean, v.w - mean};
    float q = d.x * d.x + d.y * d.y + d.z * d.z + d.w * d.w;
#pragma unroll
    for (int off = 16; off; off >>= 1) q += __shfl_xor(q, off);
    const float inv = rsqrtf(q * (1.0f / N_MAP) + EPSV);
    const float4 w4 = *(const float4*)(w + lane * 4);
    const float4 b4 = *(const float4*)(b + lane * 4);
    const float4 r4 = *(const float4*)(res + row * N_MAP + lane * 4);
    float4 o;
    o.x = fmaxf(d.x * inv * w4.x + b4.x + r4.x, 0.0f);
    o.y = fmaxf(d.y * inv * w4.y + b4.y + r4.y, 0.0f);
    o.z = fmaxf(d.z * inv * w4.z + b4.z + r4.z, 0.0f);
    o.w = fmaxf(d.w * inv * w4.w + b4.w + r4.w, 0.0f);
    *(float4*)(out + row * N_MAP + lane * 4) = o;
}

// ---------------------------------------------------------------------------
extern "C" void kernel_launch(void* const* d_in, const int* in_sizes, int n_in,
                              void* d_out, int out_size, void* d_ws, size_t ws_size,
                              hipStream_t stream) {
    const float* feat0   = (const float*)d_in[0];
    const float* W_ctr   = (const float*)d_in[1];   // [4,128,128]
    const float* W_pre   = (const float*)d_in[2];   // [4,6,128,128]
    const float* W_suc   = (const float*)d_in[3];   // [4,6,128,128]
    const float* W_left  = (const float*)d_in[4];   // [4,128,128]
    const float* W_right = (const float*)d_in[5];   // [4,128,128]
    const float* norm_w  = (const float*)d_in[6];   // [4,128]
    const float* norm_b  = (const float*)d_in[7];
    const float* W_ctr2  = (const float*)d_in[8];   // [4,128,128]
    const float* gn2_w   = (const float*)d_in[9];
    const float* gn2_b   = (const float*)d_in[10];
    const int* pre_u   = (const int*)d_in[11];
    const int* pre_v   = (const int*)d_in[12];
    const int* suc_u   = (const int*)d_in[13];
    const int* suc_v   = (const int*)d_in[14];
    const int* left_u  = (const int*)d_in[15];
    const int* left_v  = (const int*)d_in[16];
    const int* right_u = (const int*)d_in[17];
    const int* right_v = (const int*)d_in[18];

    const size_t FS = (size_t)N_NODES * N_MAP;
    float* temp = (float*)d_ws;       // accumulator; reused as second GEMM output
    float* xbuf = temp + FS;
    float* bufA = xbuf + FS;
    float* bufB = bufA + FS;

    const int GEMM_BLOCKS = N_NODES / (16 * T_PER_BLK);                    // 625
    const int EDGE_BLOCKS = 12 * (E_SCALE / 16 / T_PER_BLK)
                          + 2 * (E_LR / 16 / T_PER_BLK);                   // 7750
    const int EP_BLOCKS   = (N_NODES + 7) / 8;                             // 12500

    const float* fin = feat0;
    for (int i = 0; i < 4; ++i) {
        const size_t wOff  = (size_t)i * N_MAP * N_MAP;
        const size_t wsOff = (size_t)i * NUM_SCALES * N_MAP * N_MAP;
        float* fout = (i == 3) ? (float*)d_out : ((i & 1) ? bufB : bufA);

        // temp = feat @ W_ctr^T
        gemm_nt_kernel<<<GEMM_BLOCKS, 256, 0, stream>>>(temp, fin, W_ctr + wOff);
        // temp += scatter-add of all edge transforms
        edge_kernel<<<EDGE_BLOCKS, 256, 0, stream>>>(
            temp, fin, W_pre + wsOff, W_suc + wsOff, W_left + wOff, W_right + wOff,
            pre_u, pre_v, suc_u, suc_v, left_u, left_v, right_u, right_v);
        // x = relu(GN(temp))
        gn_relu_kernel<<<EP_BLOCKS, 256, 0, stream>>>(xbuf, temp,
                                                      norm_w + (size_t)i * N_MAP,
                                                      norm_b + (size_t)i * N_MAP);
        // temp = x @ W_ctr2^T
        gemm_nt_kernel<<<GEMM_BLOCKS, 256, 0, stream>>>(temp, xbuf, W_ctr2 + wOff);
        // feat' = relu(GN(temp) + res)
        gn_res_relu_kernel<<<EP_BLOCKS, 256, 0, stream>>>(fout, temp, fin,
                                                          gn2_w + (size_t)i * N_MAP,
                                                          gn2_b + (size_t)i * N_MAP);
        fin = fout;
    }
}